// STGCNBlock_2662879724364
// MI455X (gfx1250) — compile-verified
//
#include <hip/hip_runtime.h>
#include <hip/hip_bf16.h>
#include <math.h>

typedef __attribute__((ext_vector_type(2))) float v2f;
typedef __attribute__((ext_vector_type(8))) float v8f;

#define B_   32
#define C_   64
#define T_   512
#define V_   14
#define MID_ 8
#define TT   32            // output t per block
#define TH   (TT + 12)     // 44 (with +/-6 halo)
#define PH   (TH * V_)     // 616 halo positions
#define PHP  624           // padded to multiple of 16
#define PO   (TT * V_)     // 448 output positions (28 * 16)
#define NT1  (PHP / 16)    // 39
#define NT2  (PO  / 16)    // 28
#define EPSF 1e-5f

// workspace float offsets
#define WS_AF   0
#define WS_COMB 256
#define WS_BNSC 1152
#define WS_BNSH 1216
#define WS_MEAN 1280
#define WS_RSTD 1312

// ---------------------------------------------------------------------------
// Kernel 1: tiny prep — A_final, combined depthwise taps, BN scale/shift
// ---------------------------------------------------------------------------
__global__ void stgcn_prep(const float* __restrict__ A, const float* __restrict__ Bp,
                           const float* __restrict__ alpha_p,
                           const float* __restrict__ dw5, const float* __restrict__ dw9,
                           const float* __restrict__ dw13,
                           const float* __restrict__ bn_g, const float* __restrict__ bn_b,
                           const float* __restrict__ bn_mean, const float* __restrict__ bn_var,
                           float* __restrict__ ws) {
  __shared__ float sAt[V_ * V_];
  __shared__ float sAn[V_ * V_];
  __shared__ float sdis[V_];
  const int tid = threadIdx.x;

  if (tid < V_ * V_) {
    int i = tid / V_, j = tid % V_;
    sAt[tid] = A[tid] + (i == j ? 1.0f : 0.0f);
  }
  __syncthreads();
  if (tid < V_) {
    float deg = 0.0f;
    for (int j = 0; j < V_; ++j) deg += sAt[tid * V_ + j];
    deg = fmaxf(deg, 1.0f);
    sdis[tid] = 1.0f / sqrtf(deg);
  }
  __syncthreads();
  if (tid < V_ * V_) {
    int i = tid / V_, j = tid % V_;
    sAn[tid] = sAt[tid] * sdis[i] * sdis[j];
  }
  __syncthreads();
  if (tid < V_ * V_) {
    int i = tid / V_, j = tid % V_;
    float a2 = 0.0f;
    for (int k = 0; k < V_; ++k) a2 += sAn[i * V_ + k] * sAn[k * V_ + j];
    float amulti = sAn[tid] + 0.5f * a2;
    // softmax over row i of Bp
    float m = -1e30f;
    for (int k = 0; k < V_; ++k) m = fmaxf(m, Bp[i * V_ + k]);
    float s = 0.0f;
    for (int k = 0; k < V_; ++k) s += expf(Bp[i * V_ + k] - m);
    float bn = expf(Bp[tid] - m) / s;
    ws[WS_AF + tid] = amulti + alpha_p[0] * bn;
  }
  // combined 13-tap depthwise kernel: dw13 + centered dw9 + centered dw5
  for (int idx = tid; idx < C_ * 13; idx += blockDim.x) {
    int c = idx / 13, j = idx % 13;
    float v = dw13[c * 13 + j];
    if (j >= 2 && j <= 10) v += dw9[c * 9 + (j - 2)];
    if (j >= 4 && j <= 8)  v += dw5[c * 5 + (j - 4)];
    ws[WS_COMB + idx] = v;
  }
  if (tid < C_) {
    float sc = bn_g[tid] / sqrtf(bn_var[tid] + EPSF);
    ws[WS_BNSC + tid] = sc;
    ws[WS_BNSH + tid] = bn_b[tid] - bn_mean[tid] * sc;
  }
}

// ---------------------------------------------------------------------------
// Kernel 2: GroupNorm(1,C) stats per sample: mean, rstd over C*T*V
// ---------------------------------------------------------------------------
__global__ void stgcn_stats(const float* __restrict__ x, float* __restrict__ ws) {
  __shared__ double ssum[256];
  __shared__ double ssq[256];
  const int b = blockIdx.x;
  const int tid = threadIdx.x;
  const float* xb = x + (size_t)b * (C_ * T_ * V_);
  double s = 0.0, q = 0.0;
  for (int i = tid; i < C_ * T_ * V_; i += 256) {
    double v = (double)xb[i];
    s += v; q += v * v;
  }
  ssum[tid] = s; ssq[tid] = q;
  __syncthreads();
  for (int off = 128; off > 0; off >>= 1) {
    if (tid < off) { ssum[tid] += ssum[tid + off]; ssq[tid] += ssq[tid + off]; }
    __syncthreads();
  }
  if (tid == 0) {
    double n   = (double)(C_ * T_ * V_);
    double mu  = ssum[0] / n;
    double var = ssq[0] / n - mu * mu;
    ws[WS_MEAN + b] = (float)mu;
    ws[WS_RSTD + b] = (float)(1.0 / sqrt(var + 1e-5));
  }
}

// ---------------------------------------------------------------------------
// Kernel 3: fully fused main kernel. One block = one (b, 32-t chunk).
// GEMM B operands (xn, dwout) are stored K-pair interleaved: [k/2][p][2]
// so each WMMA B fragment is one aligned ds_load_b64.
// regA: xn(paired) -> g(row-major) -> y(row-major)
// regB: h(row-major) -> dwout(paired) -> att/a1/a2
// ---------------------------------------------------------------------------
__launch_bounds__(256)
__global__ void stgcn_main(const float* __restrict__ x,
                           const float* __restrict__ ln_g, const float* __restrict__ ln_b,
                           const float* __restrict__ w1,   const float* __restrict__ b1,
                           const float* __restrict__ pw,
                           const float* __restrict__ t1w,  const float* __restrict__ t1b,
                           const float* __restrict__ t2w,  const float* __restrict__ t2b,
                           const float* __restrict__ ws,
                           float* __restrict__ out) {
  __shared__ __align__(16) float regA[C_ * PHP];
  __shared__ __align__(16) float regB[C_ * PHP];
  __shared__ float sAf[V_ * V_];
  __shared__ float sComb[C_ * 13];
  __shared__ float sBnSc[C_];
  __shared__ float sBnSh[C_];

  const int tid  = threadIdx.x;
  const int wid  = tid >> 5;
  const int lane = tid & 31;
  const int hi   = lane >> 4;    // 0: lanes 0-15, 1: lanes 16-31
  const int l16  = lane & 15;

  const int b  = blockIdx.x / (T_ / TT);
  const int tc = blockIdx.x % (T_ / TT);
  const int t0 = tc * TT;

  for (int i = tid; i < V_ * V_; i += 256) sAf[i]   = ws[WS_AF + i];
  for (int i = tid; i < C_ * 13; i += 256) sComb[i] = ws[WS_COMB + i];
  if (tid < C_) { sBnSc[tid] = ws[WS_BNSC + tid]; sBnSh[tid] = ws[WS_BNSH + tid]; }
  const float mu = ws[WS_MEAN + b];
  const float rs = ws[WS_RSTD + b];

  // halo-position validity range: t = t0 - 6 + p/14 in [0, T_)  <=>  p in [plo, phi)
  const int plo = (6 - t0) > 0 ? (6 - t0) * V_ : 0;
  const int phi_raw = (T_ + 6 - t0) * V_;
  const int phi = phi_raw < PH ? phi_raw : PH;

  const float* xb = x + (size_t)b * (C_ * T_ * V_);
  const int goff = (t0 - 6) * V_;   // global channel-row offset: element p -> goff + p

  // ---- phase 1: load + GroupNorm -> regA paired layout [c/2][PHP][2] ----
  for (int cp = 0; cp < C_ / 2; ++cp) {
    const int c0 = 2 * cp, c1 = c0 + 1;
    const float* x0 = xb + (size_t)c0 * (T_ * V_) + goff;
    const float* x1 = x0 + (T_ * V_);
    const float s0 = rs * ln_g[c0], o0 = ln_b[c0] - mu * rs * ln_g[c0];
    const float s1 = rs * ln_g[c1], o1 = ln_b[c1] - mu * rs * ln_g[c1];
    float* dst = &regA[cp * (2 * PHP)];
    for (int p = tid; p < PHP; p += 256) {
      v2f pr; pr.x = 0.0f; pr.y = 0.0f;
      if (p >= plo && p < phi) {
        pr.x = x0[p] * s0 + o0;
        pr.y = x1[p] * s1 + o1;
      }
      *(v2f*)&dst[2 * p] = pr;
    }
  }
  __syncthreads();

  // ---- phase 2: GEMM1 h = w1 * xn + b1  (regA paired -> regB row-major) ----
  for (int tile = wid; tile < 4 * NT1; tile += 8) {
    int m = tile % 4, n = tile / 4;
    v8f acc;
#pragma unroll
    for (int r = 0; r < 8; ++r) acc[r] = b1[m * 16 + r + 8 * hi];
    const int p = n * 16 + l16;
    const int o = m * 16 + l16;
#pragma unroll
    for (int kk = 0; kk < 16; ++kk) {
      int k = kk * 4 + 2 * hi;                           // even; pair index k>>1
      v2f a  = *(const v2f*)&w1[o * C_ + k];             // global_load_b64
      v2f bb = *(const v2f*)&regA[(k >> 1) * (2 * PHP) + 2 * p];  // ds_load_b64
      acc = __builtin_amdgcn_wmma_f32_16x16x4_f32(false, a, false, bb,
                                                  (short)0, acc, false, false);
    }
#pragma unroll
    for (int r = 0; r < 8; ++r)
      regB[(m * 16 + r + 8 * hi) * PHP + p] = acc[r];
  }
  __syncthreads();

  // ---- phase 3: graph mix g = relu(h @ A_final), zero for t OOB (regB -> regA row-major) ----
  for (int c = 0; c < C_; ++c) {
    const float* hrow = &regB[c * PHP];
    float* grow = &regA[c * PHP];
    for (int p = tid; p < PH; p += 256) {
      float gv = 0.0f;
      if (p >= plo && p < phi) {
        int v = p % V_;
        const float* hr = hrow + (p - v);                // row start th*14
        float s = 0.0f;
#pragma unroll
        for (int w = 0; w < V_; ++w) s += hr[w] * sAf[w * V_ + v];
        gv = fmaxf(s, 0.0f);
      }
      grow[p] = gv;
    }
  }
  __syncthreads();

  // ---- phase 4: combined 13-tap depthwise conv -> regB paired [c/2][PO][2] ----
  for (int cp = 0; cp < C_ / 2; ++cp) {
    const int c0 = 2 * cp, c1 = c0 + 1;
    const float* g0r = &regA[c0 * PHP];
    const float* g1r = &regA[c1 * PHP];
    const float* k0 = &sComb[c0 * 13];
    const float* k1 = &sComb[c1 * 13];
    float* dst = &regB[cp * (2 * PO)];
    for (int q = tid; q < PO; q += 256) {
      float s0 = 0.0f, s1 = 0.0f;
#pragma unroll
      for (int d = 0; d < 13; ++d) {     // taps at halo position q + d*14 (j*14+v == q)
        s0 += g0r[q + d * V_] * k0[d];
        s1 += g1r[q + d * V_] * k1[d];
      }
      v2f pr; pr.x = s0; pr.y = s1;
      *(v2f*)&dst[2 * q] = pr;
    }
  }
  __syncthreads();

  // ---- phase 5: GEMM2 y = relu(BN(pw * dw))  (regB paired -> regA row-major [c][PO]) ----
  for (int tile = wid; tile < 4 * NT2; tile += 8) {
    int m = tile % 4, n = tile / 4;
    v8f acc;
#pragma unroll
    for (int r = 0; r < 8; ++r) acc[r] = 0.0f;
    const int p = n * 16 + l16;
    const int o = m * 16 + l16;
#pragma unroll
    for (int kk = 0; kk < 16; ++kk) {
      int k = kk * 4 + 2 * hi;
      v2f a  = *(const v2f*)&pw[o * C_ + k];
      v2f bb = *(const v2f*)&regB[(k >> 1) * (2 * PO) + 2 * p];
      acc = __builtin_amdgcn_wmma_f32_16x16x4_f32(false, a, false, bb,
                                                  (short)0, acc, false, false);
    }
#pragma unroll
    for (int r = 0; r < 8; ++r) {
      int oc = m * 16 + r + 8 * hi;
      float y = acc[r] * sBnSc[oc] + sBnSh[oc];
      regA[oc * PO + p] = fmaxf(y, 0.0f);
    }
  }
  __syncthreads();

  // ---- phase 6: channel attention (carved from regB; dwout is dead) ----
  float* att = regB;           // [C][TT]
  float* a1  = regB + 2048;    // [MID][TT]
  float* a2  = regB + 2304;    // [C][TT]
  for (int idx = tid; idx < C_ * TT; idx += 256) {
    int c = idx >> 5, j = idx & 31;
    const float* yr = &regA[c * PO + j * V_];
    float s = 0.0f;
#pragma unroll
    for (int v = 0; v < V_; ++v) s += yr[v];
    att[idx] = s * (1.0f / V_);
  }
  __syncthreads();
  {  // a1 = relu(t1w @ att + t1b): MID_*TT = 256 outputs, one per thread
    int m = tid >> 5, j = tid & 31;
    float s = t1b[m];
    for (int c = 0; c < C_; ++c) s += t1w[m * C_ + c] * att[c * TT + j];
    __syncthreads();
    a1[tid] = fmaxf(s, 0.0f);
  }
  __syncthreads();
  for (int idx = tid; idx < C_ * TT; idx += 256) {
    int c = idx >> 5, j = idx & 31;
    float s = t2b[c];
#pragma unroll
    for (int m = 0; m < MID_; ++m) s += t2w[c * MID_ + m] * a1[m * TT + j];
    a2[idx] = 1.0f / (1.0f + expf(-s));
  }
  __syncthreads();

  // ---- phase 7: gate + residual + relu -> global ----
  float* ob = out + (size_t)b * (C_ * T_ * V_);
  for (int c = 0; c < C_; ++c) {
    const float* yr  = &regA[c * PO];
    const float* a2r = &a2[c * TT];
    const float* xr  = xb + (size_t)c * (T_ * V_) + t0 * V_;
    float* orow      = ob + (size_t)c * (T_ * V_) + t0 * V_;
    for (int q = tid; q < PO; q += 256) {
      int j = q / V_;
      orow[q] = fmaxf(yr[q] * a2r[j] + xr[q], 0.0f);
    }
  }
}

// ---------------------------------------------------------------------------
extern "C" void kernel_launch(void* const* d_in, const int* in_sizes, int n_in,
                              void* d_out, int out_size, void* d_ws, size_t ws_size,
                              hipStream_t stream) {
  (void)in_sizes; (void)n_in; (void)out_size; (void)ws_size;
  const float* x      = (const float*)d_in[0];
  const float* A      = (const float*)d_in[1];
  const float* ln_g   = (const float*)d_in[2];
  const float* ln_b   = (const float*)d_in[3];
  const float* w1     = (const float*)d_in[4];
  const float* b1     = (const float*)d_in[5];
  const float* Bp     = (const float*)d_in[6];
  const float* alpha  = (const float*)d_in[7];
  const float* dw5    = (const float*)d_in[8];
  const float* dw9    = (const float*)d_in[9];
  const float* dw13   = (const float*)d_in[10];
  const float* pw     = (const float*)d_in[11];
  const float* bn_g   = (const float*)d_in[12];
  const float* bn_b   = (const float*)d_in[13];
  const float* bn_mean= (const float*)d_in[14];
  const float* bn_var = (const float*)d_in[15];
  const float* t1w    = (const float*)d_in[16];
  const float* t1b    = (const float*)d_in[17];
  const float* t2w    = (const float*)d_in[18];
  const float* t2b    = (const float*)d_in[19];
  float* ws  = (float*)d_ws;
  float* out = (float*)d_out;

  stgcn_prep<<<1, 256, 0, stream>>>(A, Bp, alpha, dw5, dw9, dw13,
                                    bn_g, bn_b, bn_mean, bn_var, ws);
  stgcn_stats<<<B_, 256, 0, stream>>>(x, ws);
  stgcn_main<<<B_ * (T_ / TT), 256, 0, stream>>>(x, ln_g, ln_b, w1, b1, pw,
                                                 t1w, t1b, t2w, t2b, ws, out);
}